// HeteroConvLayerCausalCus_51058571214899
// MI455X (gfx1250) — compile-verified
//
#include <hip/hip_runtime.h>
#include <hip/hip_bf16.h>

// ---------------------------------------------------------------------------
// Problem sizes (match reference)
// ---------------------------------------------------------------------------
#define N_WORD  50000
#define N_TOPIC 2000
#define N_DOC   50000
#define D_WORD  256
#define D_TOPIC 128
#define D_OUT   128

typedef __attribute__((ext_vector_type(2))) float v2f;
typedef __attribute__((ext_vector_type(8))) float v8f;

// ---------------------------------------------------------------------------
// Exact fp32 GEMM via V_WMMA_F32_16X16X4_F32.
// C[M,128] = A[M,K] @ W[K,128] + bias[128]
// One wave (32 lanes) computes a 16x128 slab of C: 8 accumulator tiles of
// 16x16 held in 8 x v8f (64 VGPRs). A is read exactly once from memory.
//
// VGPR layouts (CDNA5 ISA 7.12.2, 32-bit 16x4 A / 4x16 B / 16x16 C):
//   A: lanes 0-15 -> M=lane, VGPR0=K0,VGPR1=K1 ; lanes 16-31 -> K2,K3
//   B: lanes 0-15 -> N=lane, VGPR0=K0,VGPR1=K1 ; lanes 16-31 -> K2,K3
//   C: VGPR r: lanes 0-15 -> M=r,  N=lane ; lanes 16-31 -> M=r+8, N=lane-16
// ---------------------------------------------------------------------------
__global__ __launch_bounds__(32) void gemm_wmma_f32(
    const float* __restrict__ A, const float* __restrict__ W,
    const float* __restrict__ bias, float* __restrict__ C,
    int M, int K)
{
    const int lane = threadIdx.x & 31;
    const int half = lane >> 4;     // 0 for lanes 0-15, 1 for lanes 16-31
    const int l16  = lane & 15;
    const int mBase = blockIdx.x * 16;

    v8f acc[8] = {};                // zero-init all 8 N-tiles

    const float* __restrict__ Arow = A + (size_t)(mBase + l16) * (size_t)K;

    for (int k0 = 0; k0 < K; k0 += 4) {
        v2f a;
        a.x = Arow[k0 + 2 * half + 0];
        a.y = Arow[k0 + 2 * half + 1];
#pragma unroll
        for (int nt = 0; nt < 8; ++nt) {
            v2f b;
            b.x = W[(size_t)(k0 + 2 * half + 0) * D_OUT + nt * 16 + l16];
            b.y = W[(size_t)(k0 + 2 * half + 1) * D_OUT + nt * 16 + l16];
            acc[nt] = __builtin_amdgcn_wmma_f32_16x16x4_f32(
                /*neg_a=*/false, a, /*neg_b=*/false, b,
                /*c_mod=*/(short)0, acc[nt],
                /*reuse_a=*/false, /*reuse_b=*/false);
        }
    }

#pragma unroll
    for (int nt = 0; nt < 8; ++nt) {
        const int n  = nt * 16 + l16;
        const float bv = bias[n];
#pragma unroll
        for (int r = 0; r < 8; ++r) {
            const int mr = mBase + r + 8 * half;
            C[(size_t)mr * D_OUT + n] = acc[nt][r] + bv;
        }
    }
}

// ---------------------------------------------------------------------------
// Edge scatter: one thread per (edge, dim). acc[dst,d] += Wh[src,d]*w ;
// deg[dst] += 1 (dim 0 only). Atomics land in L2 (working set < 192MB).
// ---------------------------------------------------------------------------
__global__ void scatter_mean_acc(
    const float* __restrict__ Wh, const int* __restrict__ src,
    const int* __restrict__ dst, const float* __restrict__ w,
    float* __restrict__ acc, float* __restrict__ deg, int nE)
{
    long long idx   = (long long)blockIdx.x * blockDim.x + threadIdx.x;
    long long total = (long long)nE * D_OUT;
    if (idx >= total) return;
    const int e = (int)(idx >> 7);
    const int d = (int)(idx & (D_OUT - 1));
    const int s = src[e];
    const int t = dst[e];
    const float val = Wh[(size_t)s * D_OUT + d] * w[e];
    atomicAdd(&acc[(size_t)t * D_OUT + d], val);
    if (d == 0) atomicAdd(&deg[t], 1.0f);
}

// ---------------------------------------------------------------------------
// Finalize: mean (sum/deg, 0 if deg==0); two-relation variant sums means.
// ---------------------------------------------------------------------------
__global__ void finalize1(const float* __restrict__ acc,
                          const float* __restrict__ deg,
                          float* __restrict__ out, int n)
{
    int idx = blockIdx.x * blockDim.x + threadIdx.x;
    if (idx >= n) return;
    const float s  = acc[idx];
    const float dg = deg[idx >> 7];
    out[idx] = (dg > 0.0f) ? (s / dg) : 0.0f;
}

__global__ void finalize2(const float* __restrict__ accA,
                          const float* __restrict__ degA,
                          const float* __restrict__ accB,
                          const float* __restrict__ degB,
                          float* __restrict__ out, int n)
{
    int idx = blockIdx.x * blockDim.x + threadIdx.x;
    if (idx >= n) return;
    const int node = idx >> 7;
    const float dA = degA[node];
    const float dB = degB[node];
    const float a  = (dA > 0.0f) ? (accA[idx] / dA) : 0.0f;
    const float b  = (dB > 0.0f) ? (accB[idx] / dB) : 0.0f;
    out[idx] = a + b;
}

// ---------------------------------------------------------------------------
// Launch
// ---------------------------------------------------------------------------
extern "C" void kernel_launch(void* const* d_in, const int* in_sizes, int n_in,
                              void* d_out, int out_size, void* d_ws, size_t ws_size,
                              hipStream_t stream)
{
    // ---- inputs (setup_inputs dict order) ----
    const float* feat_word  = (const float*)d_in[0];
    const float* feat_topic = (const float*)d_in[1];
    const int *ww_src = (const int*)d_in[2],  *ww_dst = (const int*)d_in[3];
    const float *ww_w = (const float*)d_in[4];
    const int *wt_src = (const int*)d_in[5],  *wt_dst = (const int*)d_in[6];
    const float *wt_w = (const float*)d_in[7];
    const int *wd_src = (const int*)d_in[8],  *wd_dst = (const int*)d_in[9];
    const float *wd_w = (const float*)d_in[10];
    const int *td_src = (const int*)d_in[11], *td_dst = (const int*)d_in[12];
    const float *td_w = (const float*)d_in[13];
    const int *tt_src = (const int*)d_in[14], *tt_dst = (const int*)d_in[15];
    const float *tt_w = (const float*)d_in[16];
    const float *W_ww = (const float*)d_in[17], *b_ww = (const float*)d_in[18];
    const float *W_wt = (const float*)d_in[19], *b_wt = (const float*)d_in[20];
    const float *W_wd = (const float*)d_in[21], *b_wd = (const float*)d_in[22];
    const float *W_td = (const float*)d_in[23], *b_td = (const float*)d_in[24];
    const float *W_tt = (const float*)d_in[25], *b_tt = (const float*)d_in[26];

    const int E_ww = in_sizes[2], E_wt = in_sizes[5], E_wd = in_sizes[8];
    const int E_td = in_sizes[11], E_tt = in_sizes[14];

    // ---- workspace layout (floats) ----
    float* ws = (float*)d_ws;
    const size_t SZ_W = (size_t)N_WORD * D_OUT;   // 6,400,000
    const size_t SZ_T = (size_t)N_TOPIC * D_OUT;  //   256,000
    const size_t SZ_D = (size_t)N_DOC * D_OUT;    // 6,400,000

    float* Wh_ww = ws;                         // projections
    float* Wh_wt = Wh_ww + SZ_W;
    float* Wh_wd = Wh_wt + SZ_W;
    float* Wh_td = Wh_wd + SZ_W;
    float* Wh_tt = Wh_td + SZ_T;
    float* acc_ww = Wh_tt + SZ_T;              // accumulators (zeroed)
    float* acc_wt = acc_ww + SZ_W;
    float* acc_tt = acc_wt + SZ_T;
    float* acc_wd = acc_tt + SZ_T;
    float* acc_td = acc_wd + SZ_D;
    float* deg_ww = acc_td + SZ_D;             // degrees (zeroed)
    float* deg_wt = deg_ww + N_WORD;
    float* deg_tt = deg_wt + N_TOPIC;
    float* deg_wd = deg_tt + N_TOPIC;
    float* deg_td = deg_wd + N_DOC;
    float* ws_end = deg_td + N_DOC;

    // zero accumulators + degrees (contiguous region)
    hipMemsetAsync(acc_ww, 0, (size_t)(ws_end - acc_ww) * sizeof(float), stream);

    // ---- phase 1: WMMA projections ----
    gemm_wmma_f32<<<N_WORD / 16, 32, 0, stream>>>(feat_word, W_ww, b_ww, Wh_ww, N_WORD, D_WORD);
    gemm_wmma_f32<<<N_WORD / 16, 32, 0, stream>>>(feat_word, W_wt, b_wt, Wh_wt, N_WORD, D_WORD);
    gemm_wmma_f32<<<N_WORD / 16, 32, 0, stream>>>(feat_word, W_wd, b_wd, Wh_wd, N_WORD, D_WORD);
    gemm_wmma_f32<<<N_TOPIC / 16, 32, 0, stream>>>(feat_topic, W_td, b_td, Wh_td, N_TOPIC, D_TOPIC);
    gemm_wmma_f32<<<N_TOPIC / 16, 32, 0, stream>>>(feat_topic, W_tt, b_tt, Wh_tt, N_TOPIC, D_TOPIC);

    // ---- phase 2: per-relation scatter (atomic sum + degree) ----
    const int TB = 256;
    auto nblk = [](long long work, int tb) { return (unsigned)((work + tb - 1) / tb); };
    scatter_mean_acc<<<nblk((long long)E_ww * D_OUT, TB), TB, 0, stream>>>(
        Wh_ww, ww_src, ww_dst, ww_w, acc_ww, deg_ww, E_ww);
    scatter_mean_acc<<<nblk((long long)E_wt * D_OUT, TB), TB, 0, stream>>>(
        Wh_wt, wt_src, wt_dst, wt_w, acc_wt, deg_wt, E_wt);
    scatter_mean_acc<<<nblk((long long)E_tt * D_OUT, TB), TB, 0, stream>>>(
        Wh_tt, tt_src, tt_dst, tt_w, acc_tt, deg_tt, E_tt);
    scatter_mean_acc<<<nblk((long long)E_wd * D_OUT, TB), TB, 0, stream>>>(
        Wh_wd, wd_src, wd_dst, wd_w, acc_wd, deg_wd, E_wd);
    scatter_mean_acc<<<nblk((long long)E_td * D_OUT, TB), TB, 0, stream>>>(
        Wh_td, td_src, td_dst, td_w, acc_td, deg_td, E_td);

    // ---- phase 3: finalize means + cross-relation sums into d_out ----
    float* out = (float*)d_out;
    float* out_word  = out;                 // [N_WORD, 128]
    float* out_topic = out_word + SZ_W;     // [N_TOPIC,128]
    float* out_doc   = out_topic + SZ_T;    // [N_DOC,  128]

    finalize1<<<nblk((long long)SZ_W, TB), TB, 0, stream>>>(acc_ww, deg_ww, out_word, (int)SZ_W);
    finalize2<<<nblk((long long)SZ_T, TB), TB, 0, stream>>>(acc_wt, deg_wt, acc_tt, deg_tt,
                                                           out_topic, (int)SZ_T);
    finalize2<<<nblk((long long)SZ_D, TB), TB, 0, stream>>>(acc_wd, deg_wd, acc_td, deg_td,
                                                           out_doc, (int)SZ_D);
}